// OPE_82961588289680
// MI455X (gfx1250) — compile-verified
//
#include <hip/hip_runtime.h>
#include <hip/hip_bf16.h>

// ---------------------------------------------------------------------------
// Graph scatter:  out[i,d] = sum_e sqrt(w_e)*sign(x[j_e,d]-x[i_e,d]) / sum_e sqrt(w_e)
// Memory-bound (0.25 FLOP/byte). Strategy:
//   * accumulators + signal stay L2-resident (192 MB L2 >> 26 MB working set)
//   * wave32: one edge per wave, lane = feature dim -> 128B-coalesced atomics
//   * edge arrays streamed HBM->LDS via Tensor Data Mover (double buffered,
//     s_wait_tensorcnt), overlapping DMA with gather/atomic work
//   * denominator atomics: one edge per LANE (32x fewer issue slots)
//   * inner gather loop unrolled for memory-level parallelism
// ---------------------------------------------------------------------------

#define D_FEAT 32
#define CHUNK  1024          // edges staged per TDM transfer
#define BLOCK  256           // 8 waves
#define WAVES_PER_BLOCK (BLOCK / 32)
#define EDGES_PER_WAVE  (CHUNK / WAVES_PER_BLOCK)

typedef unsigned int       u32;
typedef unsigned long long u64;
typedef __attribute__((ext_vector_type(4))) u32 v4u;
typedef __attribute__((ext_vector_type(4))) int v4i;
typedef __attribute__((ext_vector_type(8))) int v8i;

// Issue one TDM 1-row tile load: global (remaining_elems bounds the tensor for
// OOB zero-fill) -> LDS byte offset. dsz: 0=1B,1=2B,2=4B,3=8B element size.
__device__ __forceinline__ void tdm_load_1d(u32 lds_byte_off, u64 gaddr,
                                            u32 remaining_elems, u32 tile_elems,
                                            u32 dsz) {
  // D# group 0: count=1, lds_addr, global_addr[56:0], type=2 ("image")
  v4u g0;
  g0.x = 1u;
  g0.y = lds_byte_off;
  g0.z = (u32)gaddr;
  g0.w = ((u32)(gaddr >> 32) & 0x1FFFFFFu) | (2u << 30);
  // D# group 1: data_size, tensor_dim0 = remaining, tensor_dim1 = 1,
  //             tile_dim0 = tile_elems, tile_dim1 = 1, strides benign.
  v8i g1;
  g1[0] = (int)(dsz << 16);                                   // workgroup_mask=0, data_size
  g1[1] = (int)((remaining_elems & 0xFFFFu) << 16);           // tensor_dim0[15:0]
  g1[2] = (int)(((remaining_elems >> 16) & 0xFFFFu) | (1u << 16)); // dim0[31:16], dim1=1
  g1[3] = (int)((tile_elems & 0xFFFFu) << 16);                // dim1 hi=0, tile_dim0
  g1[4] = 1;                                                  // tile_dim1=1, tile_dim2=0
  g1[5] = (int)tile_elems;                                    // tensor_dim0_stride lo
  g1[6] = 0;
  g1[7] = 0;
  v4i z4 = {0, 0, 0, 0};
#if defined(__clang_major__) && (__clang_major__ >= 23)
  v8i z8 = {0, 0, 0, 0, 0, 0, 0, 0};
  __builtin_amdgcn_tensor_load_to_lds(g0, g1, z4, z4, z8, 0);
#else
  __builtin_amdgcn_tensor_load_to_lds(g0, g1, z4, z4, 0);
#endif
}

__device__ __forceinline__ u32 lds_off(const void* p) {
  // generic (flat) LDS address: low 32 bits are the LDS byte offset
  return (u32)(u64)p;
}

__global__ __launch_bounds__(BLOCK) void ope_scatter_kernel(
    const float* __restrict__ signal,         // [N, 32]
    const float* __restrict__ edge_w,         // [E]
    const long long* __restrict__ edge_index, // [2, E]: row0 = j(src), row1 = i(dst)
    float* __restrict__ num,                  // [N, 32] accumulator (pre-zeroed)
    float* __restrict__ den,                  // [N]     accumulator (pre-zeroed)
    long long E) {
  __shared__ long long s_j[2][CHUNK];
  __shared__ long long s_i[2][CHUNK];
  __shared__ float     s_w[2][CHUNK];

  const int  lane = threadIdx.x & 31;
  const int  wv   = threadIdx.x >> 5;
  const long long nChunks = (E + CHUNK - 1) / CHUNK;
  const long long c0 = blockIdx.x;

  // Prologue: wave 0 stages the first chunk.
  if (c0 < nChunks && threadIdx.x < 32) {
    const long long s = c0 * CHUNK;
    const u32 rem = (u32)(E - s);
    tdm_load_1d(lds_off(&s_j[0][0]), (u64)(const void*)(edge_index + s),     rem, CHUNK, 3);
    tdm_load_1d(lds_off(&s_i[0][0]), (u64)(const void*)(edge_index + E + s), rem, CHUNK, 3);
    tdm_load_1d(lds_off(&s_w[0][0]), (u64)(const void*)(edge_w + s),         rem, CHUNK, 2);
  }

  int it = 0;
  for (long long c = c0; c < nChunks; c += gridDim.x, ++it) {
    const int cur = it & 1;
    const int nxt = cur ^ 1;
    const long long cn = c + gridDim.x;
    const bool has_next = (cn < nChunks);

    // Issue next chunk's DMA before consuming the current one.
    if (has_next && threadIdx.x < 32) {
      const long long s = cn * CHUNK;
      const u32 rem = (u32)(E - s);
      tdm_load_1d(lds_off(&s_j[nxt][0]), (u64)(const void*)(edge_index + s),     rem, CHUNK, 3);
      tdm_load_1d(lds_off(&s_i[nxt][0]), (u64)(const void*)(edge_index + E + s), rem, CHUNK, 3);
      tdm_load_1d(lds_off(&s_w[nxt][0]), (u64)(const void*)(edge_w + s),         rem, CHUNK, 2);
    }
    // All waves wait (counter is 0 for non-issuing waves -> immediate).
    if (has_next) __builtin_amdgcn_s_wait_tensorcnt(3);
    else          __builtin_amdgcn_s_wait_tensorcnt(0);
    __syncthreads();   // staged chunk visible to all waves

    // Uniform per-wave trip count (enables unrolling; TDM zero-fills OOB too).
    const long long chunk_base = c * CHUNK;
    const int ebase = wv * EDGES_PER_WAVE;
    long long avail = E - chunk_base - (long long)ebase;
    int cnt = (avail <= 0) ? 0
            : (avail < EDGES_PER_WAVE ? (int)avail : EDGES_PER_WAVE);

    // Denominator: one edge per lane -> 32 parallel scalar atomics per issue.
    for (int k = lane; k < cnt; k += 32) {
      atomicAdd(&den[(int)s_i[cur][ebase + k]], sqrtf(s_w[cur][ebase + k]));
    }

    // Numerator: lane = feature dim; unroll for gather MLP.
    #pragma unroll 4
    for (int k = 0; k < cnt; ++k) {
      const int e = ebase + k;
      const int   j  = (int)s_j[cur][e];   // wave-uniform LDS broadcast
      const int   i  = (int)s_i[cur][e];
      const float sw = sqrtf(s_w[cur][e]);
      const float a  = signal[(size_t)j * D_FEAT + lane];
      const float b  = signal[(size_t)i * D_FEAT + lane];
      const float sg = (a > b) ? 1.0f : ((a < b) ? -1.0f : 0.0f);
      atomicAdd(&num[(size_t)i * D_FEAT + lane], sw * sg);  // 32-lane coalesced line
    }
    __syncthreads();   // all waves done with buf[cur] before it is re-staged
  }
}

__global__ void ope_div_kernel(float4* __restrict__ out,
                               const float* __restrict__ den,
                               int total4) {
  const int t = blockIdx.x * blockDim.x + threadIdx.x;
  if (t < total4) {
    float4 v = out[t];
    const float inv = 1.0f / den[t >> 3];   // 8 float4 per node (D_FEAT==32)
    v.x *= inv; v.y *= inv; v.z *= inv; v.w *= inv;
    out[t] = v;
  }
}

extern "C" void kernel_launch(void* const* d_in, const int* in_sizes, int n_in,
                              void* d_out, int out_size, void* d_ws, size_t ws_size,
                              hipStream_t stream) {
  const float*     signal     = (const float*)d_in[0];      // [N,32] f32
  const float*     edge_w     = (const float*)d_in[1];      // [E,1]  f32
  const long long* edge_index = (const long long*)d_in[2];  // [2,E]  i64

  const int       N = in_sizes[0] / D_FEAT;
  const long long E = (long long)in_sizes[1];

  float* num = (float*)d_out;   // accumulate numerator directly in d_out
  float* den = (float*)d_ws;    // [N] denominator in workspace

  hipMemsetAsync(num, 0, (size_t)out_size * sizeof(float), stream);
  hipMemsetAsync(den, 0, (size_t)N * sizeof(float), stream);

  const long long nChunks = (E + CHUNK - 1) / CHUNK;
  const int grid = (nChunks < 1024) ? (int)nChunks : 1024;
  ope_scatter_kernel<<<grid, BLOCK, 0, stream>>>(signal, edge_w, edge_index,
                                                 num, den, E);

  const int total4 = (N * D_FEAT) / 4;
  ope_div_kernel<<<(total4 + 255) / 256, 256, 0, stream>>>((float4*)num, den,
                                                           total4);
}